// GraphLearningLayer_13168369730079
// MI455X (gfx1250) — compile-verified
//
#include <hip/hip_runtime.h>

typedef float v2f __attribute__((ext_vector_type(2)));
typedef float v8f __attribute__((ext_vector_type(8)));

#define B_ 8
#define N_ 384
#define D_ 256
#define H_ 1024

// ---------------------------------------------------------------------------
// WMMA GEMM (f32, 16x16x4):  C[M x NN] = act(A[M x KK] * W[NN x KK]^T + bias)
// One wave32 per 16x16 output tile. KK/NN are compile-time so the K-loop
// address math is immediate offsets (no v_mul_u64, no runtime trip guards).
//
// A fragment (16x4 f32, ISA 7.12.2): lanes 0-15 hold M=lane, K=k+0 (v0),
// k+1 (v1); lanes 16-31 hold M=lane-16, K=k+2 (v0), k+3 (v1)
//   -> contiguous float2 at A[(m0+l)*KK + k + half*2]
// B fragment (4x16): mirrored -> contiguous float2 at W[(n0+l)*KK + k + half*2]
// D (16x16 f32): VGPR r -> M = r + half*8, N = l.
// ---------------------------------------------------------------------------
template <int RELU, int KK, int NN>
__global__ void wmma_gemm_nt(const float* __restrict__ A,
                             const float* __restrict__ W,
                             const float* __restrict__ bias,
                             float* __restrict__ C)
{
    const int lane = threadIdx.x & 31;
    const int half = lane >> 4;        // 0 or 1
    const int l    = lane & 15;
    const int m0   = blockIdx.x * 16;
    const int n0   = blockIdx.y * 16;

    const float* arow = A + (size_t)(m0 + l) * KK + half * 2;
    const float* brow = W + (size_t)(n0 + l) * KK + half * 2;

    v8f acc = {};
#pragma unroll 8
    for (int k = 0; k < KK; k += 4) {
        v2f a = *(const v2f*)(arow + k);   // global_load_b64
        v2f b = *(const v2f*)(brow + k);   // global_load_b64
        acc = __builtin_amdgcn_wmma_f32_16x16x4_f32(
            /*neg_a=*/false, a, /*neg_b=*/false, b,
            /*c_mod=*/(short)0, acc, /*reuse_a=*/false, /*reuse_b=*/false);
    }

    const float bv = bias[n0 + l];
#pragma unroll
    for (int r = 0; r < 8; ++r) {
        const int row = m0 + r + half * 8;
        float v = acc[r] + bv;
        if (RELU) v = fmaxf(v, 0.0f);
        C[(size_t)row * NN + (n0 + l)] = v;
    }
}

// ---------------------------------------------------------------------------
// Pairwise weighted L1 distance + mask + leaky_relu.
// Block = 256 threads = one 16x16 (i,j) tile for one batch b.
// LDS-staged row tiles, pad = 260 floats (16B aligned, rotates banks by 4
// per row so the 16 xj rows hit distinct banks).
// ---------------------------------------------------------------------------
#define PAD 260
__global__ void l1dist_kernel(const float* __restrict__ xhat,
                              const float* __restrict__ w,
                              const long long* __restrict__ box_num,
                              float* __restrict__ dist)
{
    __shared__ float xi[16][PAD];
    __shared__ float xj[16][PAD];
    __shared__ float wsh[D_];
    __shared__ float swsh;

    const int b  = blockIdx.z;
    const int i0 = blockIdx.y * 16;
    const int j0 = blockIdx.x * 16;
    const int tid = threadIdx.x;   // 0..255

    // cooperative tile load (float4, coalesced)
#pragma unroll
    for (int idx = tid; idx < 16 * (D_ / 4); idx += 256) {
        const int row = idx >> 6;          // /(D_/4)
        const int d4  = idx & 63;
        const float4 vi = ((const float4*)(xhat + (size_t)(b * N_ + i0 + row) * D_))[d4];
        ((float4*)&xi[row][0])[d4] = vi;
        const float4 vj = ((const float4*)(xhat + (size_t)(b * N_ + j0 + row) * D_))[d4];
        ((float4*)&xj[row][0])[d4] = vj;
    }
    wsh[tid] = w[tid];                     // D_ == 256 == blockDim
    __syncthreads();
    if (tid == 0) {
        float s = 0.0f;
        for (int d = 0; d < D_; ++d) s += wsh[d];
        swsh = s;
    }
    __syncthreads();

    const int ti = tid >> 4, tj = tid & 15;
    float acc = 0.0f;
#pragma unroll 4
    for (int d = 0; d < D_; d += 4) {
        const float4 a  = *(const float4*)&xi[ti][d];   // ds_load_b128 (broadcast)
        const float4 c  = *(const float4*)&xj[tj][d];   // ds_load_b128
        const float4 ww = *(const float4*)&wsh[d];
        acc = fmaf(fabsf(a.x - c.x), ww.x, acc);
        acc = fmaf(fabsf(a.y - c.y), ww.y, acc);
        acc = fmaf(fabsf(a.z - c.z), ww.z, acc);
        acc = fmaf(fabsf(a.w - c.w), ww.w, acc);
    }

    const long long bn = box_num[b];
    const int i = i0 + ti, j = j0 + tj;
    const bool valid = ((long long)i < bn) && ((long long)j < bn);
    float dv = acc - (valid ? 0.0f : swsh);             // mask * sum(w)
    dv = dv > 0.0f ? dv : 0.01f * dv;                   // leaky_relu
    dist[(size_t)(b * N_ + i) * N_ + j] = dv;
}

// ---------------------------------------------------------------------------
// Row softmax with adj weighting, in-place on dist (soft_adj region of d_out).
// Block = 128 threads, each owning 3 of the 384 columns.
// ---------------------------------------------------------------------------
__global__ void softmax_kernel(const float* __restrict__ adj,
                               float* __restrict__ soft)
{
    __shared__ float red[128];
    const int b = blockIdx.y, i = blockIdx.x;
    const int tid = threadIdx.x;                     // 0..127
    float* row        = soft + (size_t)(b * N_ + i) * N_;
    const float* arow = adj  + (size_t)(b * N_ + i) * N_;

    const float v0 = row[tid], v1 = row[tid + 128], v2 = row[tid + 256];

    red[tid] = fmaxf(v0, fmaxf(v1, v2));
    __syncthreads();
#pragma unroll
    for (int s = 64; s > 0; s >>= 1) {
        if (tid < s) red[tid] = fmaxf(red[tid], red[tid + s]);
        __syncthreads();
    }
    const float m = red[0];
    __syncthreads();

    const float e0 = arow[tid]       * __expf(v0 - m);
    const float e1 = arow[tid + 128] * __expf(v1 - m);
    const float e2 = arow[tid + 256] * __expf(v2 - m);

    red[tid] = e0 + e1 + e2;
    __syncthreads();
#pragma unroll
    for (int s = 64; s > 0; s >>= 1) {
        if (tid < s) red[tid] += red[tid + s];
        __syncthreads();
    }
    const float inv = 1.0f / red[0];

    row[tid]       = e0 * inv + 1e-10f;
    row[tid + 128] = e1 * inv + 1e-10f;
    row[tid + 256] = e2 * inv + 1e-10f;
}

// ---------------------------------------------------------------------------
extern "C" void kernel_launch(void* const* d_in, const int* in_sizes, int n_in,
                              void* d_out, int out_size, void* d_ws, size_t ws_size,
                              hipStream_t stream)
{
    const float*     x       = (const float*)d_in[0];
    const float*     adj     = (const float*)d_in[1];
    const long long* box_num = (const long long*)d_in[2];   // int64 per reference
    const float*     fc1_w   = (const float*)d_in[3];       // (H, D)
    const float*     fc1_b   = (const float*)d_in[4];       // (H,)
    const float*     fc2_w   = (const float*)d_in[5];       // (D, H)
    const float*     fc2_b   = (const float*)d_in[6];       // (D,)
    const float*     learn_w = (const float*)d_in[7];       // (D,)

    float* soft = (float*)d_out;                            // (B, N, N)
    float* xhat = (float*)d_out + (size_t)B_ * N_ * N_;     // (B, N, D)
    float* h    = (float*)d_ws;                             // (B*N, H) = 12.6 MB

    const int M = B_ * N_;   // 3072

    // GEMM1: h = relu(x (M x D) * fc1_w (H x D)^T + fc1_b)   -> M x H
    wmma_gemm_nt<1, D_, H_><<<dim3(M / 16, H_ / 16), 32, 0, stream>>>(
        x, fc1_w, fc1_b, h);

    // GEMM2: xhat = h (M x H) * fc2_w (D x H)^T + fc2_b      -> M x D
    wmma_gemm_nt<0, H_, D_><<<dim3(M / 16, D_ / 16), 32, 0, stream>>>(
        h, fc2_w, fc2_b, xhat);

    // dist + mask + leaky_relu  -> soft_adj region of d_out
    l1dist_kernel<<<dim3(N_ / 16, N_ / 16, B_), 256, 0, stream>>>(
        xhat, learn_w, box_num, soft);

    // row softmax with adj, in place
    softmax_kernel<<<dim3(N_, B_), 128, 0, stream>>>(adj, soft);
}